// Sequence_781684047902
// MI455X (gfx1250) — compile-verified
//
#include <hip/hip_runtime.h>

#define H    51
#define G4   204      // 4*H gate width
#define GP   208      // padded gate width (13*16)
#define HP   52       // padded hidden (13*4)
#define T_IN 512
#define T_OUT 576     // 512 + 64 future
#define KC   13       // K chunks of 4 covering H=51 (padded to 52)
#define NWAVE 13      // gate-tile waves; wave 13 is the linear-head wave
#define NTHREADS 448  // 14 waves * 32

typedef __attribute__((ext_vector_type(2))) float v2f;
typedef __attribute__((ext_vector_type(8))) float v8f;

#if __has_builtin(__builtin_amdgcn_tanhf)
__device__ __forceinline__ float tanhf_(float x) {
    return __builtin_amdgcn_tanhf(x);
}
__device__ __forceinline__ float sigmoidf_(float x) {
    return __builtin_amdgcn_tanhf(0.5f * x) * 0.5f + 0.5f;
}
#else
__device__ __forceinline__ float tanhf_(float x) {
    // 1 - 2/(e^{2x}+1): overflow-safe (inf -> 1, 0 -> -1), no IEEE divide
    float e = __expf(2.0f * x);
    return 1.0f - 2.0f * __builtin_amdgcn_rcpf(e + 1.0f);
}
__device__ __forceinline__ float sigmoidf_(float x) {
    return __builtin_amdgcn_rcpf(1.0f + __expf(-x));
}
#endif

__launch_bounds__(NTHREADS, 1)
__global__ void lstm_seq_kernel(const float* __restrict__ input,
                                const float* __restrict__ Wih1,
                                const float* __restrict__ Whh1,
                                const float* __restrict__ bih1,
                                const float* __restrict__ bhh1,
                                const float* __restrict__ Wih2,
                                const float* __restrict__ Whh2,
                                const float* __restrict__ bih2,
                                const float* __restrict__ bhh2,
                                const float* __restrict__ Wl,
                                const float* __restrict__ bl,
                                float* __restrict__ out)
{
    __shared__ float Gs[16 * GP];     // gate pre-activations for the 16-row tile
    __shared__ float h1s[16 * HP];    // layer-1 hidden (col 51 stays zero)
    __shared__ float h2s[16 * HP];    // layer-2 hidden
    __shared__ float b1s[G4];
    __shared__ float b2s[G4];
    __shared__ float wih1s[G4];
    __shared__ float xs[16];          // per-row scalar input
    __shared__ float bls;

    const int tid  = threadIdx.x;
    const int wid  = tid >> 5;        // 0..12 gate tiles, 13 = head wave
    const int lane = tid & 31;
    const int hh   = lane >> 4;       // lane half (selects K pair / M+8)
    const int l16  = lane & 15;
    const int n0   = wid * 16;        // gate-column tile base
    const int b0   = blockIdx.x * 16; // batch-row tile base

    // ---- per-wave weight fragments in registers (B operand = W^T) ----
    // B 4x16 f32 layout: VGPR r, lane-half hh -> K = 4*kk + 2*hh + r, N = n0+l16
    v2f whh1f[KC], wih2f[KC], whh2f[KC];
    if (wid < NWAVE) {
        const int n = n0 + l16;
        #pragma unroll
        for (int kk = 0; kk < KC; ++kk) {
            #pragma unroll
            for (int r = 0; r < 2; ++r) {
                const int k = kk * 4 + 2 * hh + r;
                float a = 0.0f, bb = 0.0f, c = 0.0f;
                if (n < G4 && k < H) {
                    a  = Whh1[n * H + k];
                    bb = Wih2[n * H + k];
                    c  = Whh2[n * H + k];
                }
                whh1f[kk][r] = a;
                wih2f[kk][r] = bb;
                whh2f[kk][r] = c;
            }
        }
    } else {
        // head wave: Wl occupies column 0 of a zero-padded tile, stored in whh1f
        #pragma unroll
        for (int kk = 0; kk < KC; ++kk) {
            #pragma unroll
            for (int r = 0; r < 2; ++r) {
                const int k = kk * 4 + 2 * hh + r;
                whh1f[kk][r] = (l16 == 0 && k < H) ? Wl[k] : 0.0f;
            }
        }
    }

    // ---- init LDS constants and zero state ----
    if (tid < G4) {
        b1s[tid]   = bih1[tid] + bhh1[tid];
        b2s[tid]   = bih2[tid] + bhh2[tid];
        wih1s[tid] = Wih1[tid];
    }
    if (tid == 0) bls = bl[0];
    if (tid < 16) xs[tid] = input[(b0 + tid) * T_IN];
    for (int i = tid; i < 16 * HP; i += NTHREADS) { h1s[i] = 0.0f; h2s[i] = 0.0f; }

    // element-wise ownership: e = row*H + j over 816 elements, <=2 per thread;
    // cell states live in these threads' registers for the whole kernel
    const int  e1 = tid + NTHREADS;
    const int  r0 = tid / H, j0 = tid % H;
    const int  r1 = e1 / H,  j1 = e1 % H;
    const bool v1 = (e1 < 16 * H);
    float c1a = 0.0f, c1b = 0.0f, c2a = 0.0f, c2b = 0.0f;

    __syncthreads();

    for (int t = 0; t <= T_OUT; ++t) {
        // ===== phase A: waves 0..12 do layer-1 GEMM for step t,
        //                wave 13 does the linear head for step t-1 =====
        if (wid < NWAVE) {
            if (t < T_OUT) {
                v8f acc = {};
                #pragma unroll
                for (int kk = 0; kk < KC; ++kk) {
                    v2f a = *(const v2f*)&h1s[l16 * HP + kk * 4 + 2 * hh];
                    acc = __builtin_amdgcn_wmma_f32_16x16x4_f32(
                            false, a, false, whh1f[kk], (short)0, acc, false, false);
                }
                #pragma unroll
                for (int v = 0; v < 8; ++v)
                    Gs[(v + 8 * hh) * GP + n0 + l16] = acc[v];
            }
        } else if (t > 0) {
            v8f acc = {};
            #pragma unroll
            for (int kk = 0; kk < KC; ++kk) {
                v2f a = *(const v2f*)&h2s[l16 * HP + kk * 4 + 2 * hh];
                acc = __builtin_amdgcn_wmma_f32_16x16x4_f32(
                        false, a, false, whh1f[kk], (short)0, acc, false, false);
            }
            if (l16 == 0) {                // lanes 0 and 16 hold column 0
                const float blv = bls;
                #pragma unroll
                for (int v = 0; v < 8; ++v) {
                    const int   row = v + 8 * hh;
                    const float val = acc[v] + blv;
                    out[(b0 + row) * T_OUT + (t - 1)] = val;
                    if (t < T_IN) {
                        xs[row] = input[(b0 + row) * T_IN + t];
                    } else {
                        xs[row] = val;     // future phase: feed prediction back
                    }
                }
            }
        }
        if (t == T_OUT) break;
        __syncthreads();

        // ---- element-wise LSTM cell, layer 1 (adds x*Wih1 + b1) ----
        {
            const float x = xs[r0];
            const float pi = Gs[r0 * GP + j0]         + x * wih1s[j0]         + b1s[j0];
            const float pf = Gs[r0 * GP + H + j0]     + x * wih1s[H + j0]     + b1s[H + j0];
            const float pg = Gs[r0 * GP + 2 * H + j0] + x * wih1s[2 * H + j0] + b1s[2 * H + j0];
            const float po = Gs[r0 * GP + 3 * H + j0] + x * wih1s[3 * H + j0] + b1s[3 * H + j0];
            c1a = sigmoidf_(pf) * c1a + sigmoidf_(pi) * tanhf_(pg);
            h1s[r0 * HP + j0] = sigmoidf_(po) * tanhf_(c1a);
        }
        if (v1) {
            const float x = xs[r1];
            const float pi = Gs[r1 * GP + j1]         + x * wih1s[j1]         + b1s[j1];
            const float pf = Gs[r1 * GP + H + j1]     + x * wih1s[H + j1]     + b1s[H + j1];
            const float pg = Gs[r1 * GP + 2 * H + j1] + x * wih1s[2 * H + j1] + b1s[2 * H + j1];
            const float po = Gs[r1 * GP + 3 * H + j1] + x * wih1s[3 * H + j1] + b1s[3 * H + j1];
            c1b = sigmoidf_(pf) * c1b + sigmoidf_(pi) * tanhf_(pg);
            h1s[r1 * HP + j1] = sigmoidf_(po) * tanhf_(c1b);
        }
        __syncthreads();

        // ===== layer 2: G = h1 @ Wih2^T + h2 @ Whh2^T (waves 0..12) =====
        if (wid < NWAVE) {
            v8f acc = {};
            #pragma unroll
            for (int kk = 0; kk < KC; ++kk) {
                v2f a = *(const v2f*)&h1s[l16 * HP + kk * 4 + 2 * hh];
                acc = __builtin_amdgcn_wmma_f32_16x16x4_f32(
                        false, a, false, wih2f[kk], (short)0, acc, false, false);
            }
            #pragma unroll
            for (int kk = 0; kk < KC; ++kk) {
                v2f a = *(const v2f*)&h2s[l16 * HP + kk * 4 + 2 * hh];
                acc = __builtin_amdgcn_wmma_f32_16x16x4_f32(
                        false, a, false, whh2f[kk], (short)0, acc, false, false);
            }
            #pragma unroll
            for (int v = 0; v < 8; ++v)
                Gs[(v + 8 * hh) * GP + n0 + l16] = acc[v];
        }
        __syncthreads();

        // ---- element-wise LSTM cell, layer 2 ----
        {
            const float pi = Gs[r0 * GP + j0]         + b2s[j0];
            const float pf = Gs[r0 * GP + H + j0]     + b2s[H + j0];
            const float pg = Gs[r0 * GP + 2 * H + j0] + b2s[2 * H + j0];
            const float po = Gs[r0 * GP + 3 * H + j0] + b2s[3 * H + j0];
            c2a = sigmoidf_(pf) * c2a + sigmoidf_(pi) * tanhf_(pg);
            h2s[r0 * HP + j0] = sigmoidf_(po) * tanhf_(c2a);
        }
        if (v1) {
            const float pi = Gs[r1 * GP + j1]         + b2s[j1];
            const float pf = Gs[r1 * GP + H + j1]     + b2s[H + j1];
            const float pg = Gs[r1 * GP + 2 * H + j1] + b2s[2 * H + j1];
            const float po = Gs[r1 * GP + 3 * H + j1] + b2s[3 * H + j1];
            c2b = sigmoidf_(pf) * c2b + sigmoidf_(pi) * tanhf_(pg);
            h2s[r1 * HP + j1] = sigmoidf_(po) * tanhf_(c2b);
        }
        __syncthreads();
    }
}

extern "C" void kernel_launch(void* const* d_in, const int* in_sizes, int n_in,
                              void* d_out, int out_size, void* d_ws, size_t ws_size,
                              hipStream_t stream) {
    const float* input = (const float*)d_in[0];
    const float* Wih1  = (const float*)d_in[1];
    const float* Whh1  = (const float*)d_in[2];
    const float* bih1  = (const float*)d_in[3];
    const float* bhh1  = (const float*)d_in[4];
    const float* Wih2  = (const float*)d_in[5];
    const float* Whh2  = (const float*)d_in[6];
    const float* bih2  = (const float*)d_in[7];
    const float* bhh2  = (const float*)d_in[8];
    const float* Wl    = (const float*)d_in[9];
    const float* bl    = (const float*)d_in[10];

    const int B = in_sizes[0] / T_IN;   // 1024
    lstm_seq_kernel<<<dim3(B / 16), dim3(NTHREADS), 0, stream>>>(
        input, Wih1, Whh1, bih1, bhh1, Wih2, Whh2, bih2, bhh2, Wl, bl,
        (float*)d_out);
}